// MyGCN_37538014167295
// MI455X (gfx1250) — compile-verified
//
#include <hip/hip_runtime.h>

typedef __attribute__((ext_vector_type(2))) float v2f;
typedef __attribute__((ext_vector_type(8))) float v8f;

// ---------------------------------------------------------------------------
// Utility kernels
// ---------------------------------------------------------------------------
__global__ void zero_f32(float* __restrict__ p, long long n) {
    long long i = (long long)blockIdx.x * blockDim.x + threadIdx.x;
    if (i < n) p[i] = 0.0f;
}

__global__ void deg_kernel(const long long* __restrict__ row, float* __restrict__ deg,
                           long long E) {
    long long e = (long long)blockIdx.x * blockDim.x + threadIdx.x;
    if (e < E) atomicAdd(&deg[row[e]], 1.0f);
}

__global__ void dis_kernel(const float* __restrict__ deg, float* __restrict__ dis,
                           long long N) {
    long long i = (long long)blockIdx.x * blockDim.x + threadIdx.x;
    if (i < N) {
        float d = deg[i];
        dis[i] = (d > 0.0f) ? rsqrtf(d) : 0.0f;
    }
}

// One WAVE per edge; lane handles 4 consecutive features. D == 128.
// row[e]/col[e] are wave-uniform -> force into SGPRs via readfirstlane so the
// gather/scatter addresses use the scalar-base (saddr) form and each wave does
// one fully coalesced 512B load.
__global__ void scatter_edges(const float* __restrict__ src,
                              const long long* __restrict__ row,
                              const long long* __restrict__ col,
                              float* __restrict__ agg, long long E) {
    const int D = 128;
    const int wave = threadIdx.x >> 5;
    const int lane = threadIdx.x & 31;
    long long e = (long long)blockIdx.x * (blockDim.x >> 5) + wave;
    if (e >= E) return;
    int r = __builtin_amdgcn_readfirstlane((int)row[e]);  // node ids < 2^31
    int c = __builtin_amdgcn_readfirstlane((int)col[e]);
    int f = lane * 4;
    const float4 v = *(const float4*)(src + (long long)c * D + f);
    float* dst = agg + (long long)r * D + f;
    atomicAdd(dst + 0, v.x);
    atomicAdd(dst + 1, v.y);
    atomicAdd(dst + 2, v.z);
    atomicAdd(dst + 3, v.w);
}

// Repack W[k][c] (K x NOUT row-major) so one lane's B fragments for ALL
// N-tiles at one k-step are contiguous:
//   cell(kpair, n) = kpair*16 + n   (kpair = k>>1, n = c&15, t = c>>4)
//   Wp[cell*NT*2 + t*2 + (k&1)] = W[k*NOUT + c]
// -> per (kpair, n) cell: NT*2 floats = NT/2 float4s, loadable as b128s.
__global__ void pack_w(const float* __restrict__ W, float* __restrict__ Wp,
                       int K, int NOUT) {
    int idx = blockIdx.x * blockDim.x + threadIdx.x;
    if (idx >= K * NOUT) return;
    int k = idx / NOUT;
    int c = idx % NOUT;
    int t = c >> 4;
    int n = c & 15;
    int NT = NOUT >> 4;
    Wp[((size_t)((k >> 1) * 16 + n) * NT + t) * 2 + (k & 1)] = W[idx];
}

// ---------------------------------------------------------------------------
// Fused (D^-1/2 * agg) @ W + bias (+ReLU) with V_WMMA_F32_16X16X4_F32.
// W staged in LDS (padded), per-k-step fragments double-buffered in registers.
// One wave: 16-row M-tile x all NT N-tiles, K = 128.
// ---------------------------------------------------------------------------
template <int NT, bool RELU>
__global__ __launch_bounds__(256, 1) void gcn_gemm(
        const float* __restrict__ A,     // [N,128] unscaled agg
        const float* __restrict__ dis,   // [N]
        const float4* __restrict__ Wp4,  // packed weights (see pack_w)
        const float* __restrict__ bias,  // [NOUT]
        float* __restrict__ out,         // [N,NOUT]
        int nMTiles) {
    constexpr int K = 128;
    constexpr int NOUT = NT * 16;
    constexpr int NB = NT / 2;     // float4s per (kpair,n) cell
    constexpr int NBP = NB + 1;    // padded LDS stride (bank spread)
    constexpr int KPAIRS = K / 2;  // 64
    constexpr int WSTEP = 32 * NBP;  // float4s per k-step (2 kpairs * 16 cells)

    extern __shared__ float4 lds_w[];  // KPAIRS*16*NBP float4s

    // Cooperative copy: global packed W -> padded LDS (all threads, pre-barrier)
    for (int idx = threadIdx.x; idx < KPAIRS * 16 * NB; idx += blockDim.x) {
        int cell = idx / NB;
        int j = idx % NB;
        lds_w[cell * NBP + j] = Wp4[idx];
    }
    __syncthreads();

    const int wave = threadIdx.x >> 5;
    const int lane = threadIdx.x & 31;
    const int mtile = blockIdx.x * (blockDim.x >> 5) + wave;
    if (mtile >= nMTiles) return;  // wave-uniform: EXEC stays all-1s for WMMA

    const int n  = lane & 15;  // N index (B/C/D) == M index (A)
    const int kh = lane >> 4;  // K-half select

    const long long mrow = (long long)mtile * 16 + n;
    const float s = dis[mrow];  // fold D^-1/2 into A
    const float* __restrict__ arow = A + mrow * K + 2 * kh;
    const float4* __restrict__ wlane = lds_w + ((size_t)kh * 16 + n) * NBP;

    v8f acc[NT];
#pragma unroll
    for (int t = 0; t < NT; ++t)
#pragma unroll
        for (int r = 0; r < 8; ++r) acc[t][r] = 0.0f;

    // Prologue: fragment buffers for k-step 0
    float2 acur = *(const float2*)(arow);
    float4 wcur[NB];
#pragma unroll
    for (int j = 0; j < NB; ++j) wcur[j] = wlane[j];

#pragma unroll
    for (int i = 0; i < K / 4; ++i) {
        float2 an;
        float4 wn[NB];
        if (i + 1 < K / 4) {  // prefetch next k-step while WMMAs run
            an = *(const float2*)(arow + 4 * (i + 1));
#pragma unroll
            for (int j = 0; j < NB; ++j) wn[j] = wlane[(i + 1) * WSTEP + j];
        }
        v2f a;
        a.x = acur.x * s;
        a.y = acur.y * s;
#pragma unroll
        for (int t = 0; t < NT; ++t) {
            const float4 w4 = wcur[t >> 1];
            v2f b;
            if (t & 1) { b.x = w4.z; b.y = w4.w; }
            else       { b.x = w4.x; b.y = w4.y; }
            acc[t] = __builtin_amdgcn_wmma_f32_16x16x4_f32(
                false, a, false, b, (short)0, acc[t], false, false);
        }
        if (i + 1 < K / 4) {
            acur = an;
#pragma unroll
            for (int j = 0; j < NB; ++j) wcur[j] = wn[j];
        }
    }

    // Epilogue: bias (+ReLU); C/D layout: VGPR r -> row r + 8*kh, col = lane&15
#pragma unroll
    for (int t = 0; t < NT; ++t) {
        const float bv = bias[t * 16 + n];
#pragma unroll
        for (int r = 0; r < 8; ++r) {
            float v = acc[t][r] + bv;
            if (RELU) v = fmaxf(v, 0.0f);
            out[(long long)(mtile * 16 + r + 8 * kh) * NOUT + t * 16 + n] = v;
        }
    }
}

// Scalar fallback for N % 16 rows (zero blocks at this problem size).
__global__ void gemm_tail(const float* __restrict__ A, const float* __restrict__ dis,
                          const float* __restrict__ W, const float* __restrict__ bias,
                          float* __restrict__ out, long long rowStart, long long nRows,
                          int NOUT, int relu) {
    const int K = 128;
    long long idx = (long long)blockIdx.x * blockDim.x + threadIdx.x;
    if (idx >= nRows * NOUT) return;
    long long r = rowStart + idx / NOUT;
    int nc = (int)(idx % NOUT);
    float s = dis[r];
    float acc = 0.0f;
    for (int k = 0; k < K; ++k) acc += A[r * K + k] * s * W[k * NOUT + nc];
    acc += bias[nc];
    if (relu) acc = fmaxf(acc, 0.0f);
    out[r * NOUT + nc] = acc;
}

// ---------------------------------------------------------------------------
static inline unsigned cdiv_ll(long long a, long long b) {
    return (unsigned)((a + b - 1) / b);
}

extern "C" void kernel_launch(void* const* d_in, const int* in_sizes, int n_in,
                              void* d_out, int out_size, void* d_ws, size_t ws_size,
                              hipStream_t stream) {
    const int D = 128, DH = 128, DOUT = 64;

    const float*     x  = (const float*)d_in[0];
    const long long* ei = (const long long*)d_in[1];  // int64 [2,E]
    // d_in[2] = size scalar (derived from in_sizes instead)
    const float* w1 = (const float*)d_in[3];
    const float* b1 = (const float*)d_in[4];
    const float* w2 = (const float*)d_in[5];
    const float* b2 = (const float*)d_in[6];

    const long long N = in_sizes[0] / D;
    const long long E = in_sizes[1] / 2;
    const long long* row = ei;
    const long long* col = ei + E;

    // Workspace layout (256B aligned slabs)
    char* ws = (char*)d_ws;
    size_t off = 0;
    auto alloc = [&](size_t bytes) -> void* {
        void* p = ws + off;
        off += (bytes + 255) & ~(size_t)255;
        return p;
    };
    float* deg = (float*)alloc((size_t)N * 4);
    float* dis = (float*)alloc((size_t)N * 4);
    float* agg = (float*)alloc((size_t)N * D * 4);
    float* h   = (float*)alloc((size_t)N * DH * 4);
    float* w1p = (float*)alloc((size_t)D * DH * 4);
    float* w2p = (float*)alloc((size_t)DH * DOUT * 4);
    (void)ws_size;

    float* out = (float*)d_out;

    const int B = 256;
    const long long aggElems = N * D;
    const int nMT = (int)(N / 16);
    const long long rem = N % 16;

    const size_t lds1 = (size_t)64 * 16 * ((128 / 32) + 1) * sizeof(float4);  // NT=8: 80KB
    const size_t lds2 = (size_t)64 * 16 * ((64 / 32) + 1) * sizeof(float4);   // NT=4: 48KB

    // Degree + normalization (shared by both layers)
    zero_f32<<<cdiv_ll(N, B), B, 0, stream>>>(deg, N);
    deg_kernel<<<cdiv_ll(E, B), B, 0, stream>>>(row, deg, E);
    dis_kernel<<<cdiv_ll(N, B), B, 0, stream>>>(deg, dis, N);

    // Repack weights for b128 B-loads
    pack_w<<<cdiv_ll((long long)D * DH, B), B, 0, stream>>>(w1, w1p, D, DH);
    pack_w<<<cdiv_ll((long long)DH * DOUT, B), B, 0, stream>>>(w2, w2p, DH, DOUT);

    // ---- Layer 1: h = relu((D^-1/2 A x) @ w1 + b1)
    zero_f32<<<cdiv_ll(aggElems, B), B, 0, stream>>>(agg, aggElems);
    scatter_edges<<<cdiv_ll(E, B / 32), B, 0, stream>>>(x, row, col, agg, E);
    gcn_gemm<8, true><<<cdiv_ll(nMT, 8), B, lds1, stream>>>(
        agg, dis, (const float4*)w1p, b1, h, nMT);
    if (rem > 0)
        gemm_tail<<<cdiv_ll(rem * DH, B), B, 0, stream>>>(agg, dis, w1, b1, h,
                                                          (long long)nMT * 16, rem, DH, 1);

    // ---- Layer 2: out = (D^-1/2 A h) @ w2 + b2
    zero_f32<<<cdiv_ll(aggElems, B), B, 0, stream>>>(agg, aggElems);
    scatter_edges<<<cdiv_ll(E, B / 32), B, 0, stream>>>(h, row, col, agg, E);
    gcn_gemm<4, false><<<cdiv_ll(nMT, 8), B, lds2, stream>>>(
        agg, dis, (const float4*)w2p, b2, out, nMT);
    if (rem > 0)
        gemm_tail<<<cdiv_ll(rem * DOUT, B), B, 0, stream>>>(agg, dis, w2, b2, out,
                                                            (long long)nMT * 16, rem, DOUT, 0);
}